// SelectiveScanLoop_4879082848710
// MI455X (gfx1250) — compile-verified
//
#include <hip/hip_runtime.h>
#include <stdint.h>

// ---------------------------------------------------------------------------
// Selective scan: x_i = dA_i * x_{i-1} + dBu_i ;  y_i[d] = sum_n x_i[d,n]*C_i[n]
// B=1, L=2048, D=2048, N=16 (dims derived from in_sizes at launch).
//
// Memory-bound: deltaA+deltaB_u = 512MB single-use f32 stream; at 23.3TB/s the
// floor is ~22us, total FLOPs only ~0.54G => no matrix-pipe benefit possible.
// One thread per (d,n); 6-deep async global->LDS pipeline (ASYNCcnt) hides
// load latency behind the serial recurrence; TH_LOAD_NT keeps the single-use
// stream from thrashing the 192MB L2 (C stays RT: 128KB, reread by all blocks).
// ---------------------------------------------------------------------------

#define ASYNC_LDS_LOAD_NT(ldsoff, voff, sbase)                                 \
  asm volatile("global_load_async_to_lds_b32 %0, %1, %2 th:TH_LOAD_NT"        \
               :: "v"(ldsoff), "v"(voff), "s"(sbase) : "memory")

#define ASYNC_LDS_LOAD(ldsoff, voff, sbase)                                    \
  asm volatile("global_load_async_to_lds_b32 %0, %1, %2"                      \
               :: "v"(ldsoff), "v"(voff), "s"(sbase) : "memory")

__global__ __launch_bounds__(256) void SelectiveScan_async_kernel(
    const float* __restrict__ x0,    // (D*N) initial state
    const float* __restrict__ dA,    // (L, D, N)
    const float* __restrict__ dBu,   // (L, D, N)
    const float* __restrict__ Cm,    // (L, N)
    float* __restrict__ out,         // (L, D)
    int L, int DN)                   // DN = D*N
{
  constexpr int T      = 256;  // threads per block: 16 d x 16 n
  constexpr int PRE    = 6;    // prefetch distance (iterations ahead)
  constexpr int STAGES = 8;    // LDS stages (> PRE+1 => slack vs ds_load race)

  __shared__ float sA[STAGES * T];
  __shared__ float sB[STAGES * T];
  __shared__ float sC[STAGES * T];

  const int t          = threadIdx.x;
  const int n          = t & 15;                       // state index
  const int blockStart = blockIdx.x * T;               // flat (d,n) base
  const int D          = DN >> 4;
  const int dIdx       = (blockStart >> 4) + (t >> 4); // global d index

  // LDS byte offsets (flat-address low 32 bits == workgroup LDS offset)
  const uint32_t ldsA0 = (uint32_t)(uintptr_t)&sA[t];
  const uint32_t ldsB0 = (uint32_t)(uintptr_t)&sB[t];
  const uint32_t ldsC0 = (uint32_t)(uintptr_t)&sC[t];
  const uint32_t stageBytes = T * (uint32_t)sizeof(float);

  // GVS addressing: 64-bit SGPR base + per-lane 32-bit byte offset
  uint32_t       offAB  = (uint32_t)(blockStart + t) * 4u; // pos of next issue
  uint32_t       offC   = (uint32_t)n * 4u;
  const uint32_t stepAB = (uint32_t)DN * 4u;

  float xs = x0[blockStart + t];                       // carried state

  // ---- prime the pipeline: stages 0..PRE-1 (assume L > PRE; L=2048) ------
#pragma unroll
  for (int j = 0; j < PRE; ++j) {
    const uint32_t s = (uint32_t)j * stageBytes;
    ASYNC_LDS_LOAD_NT(ldsA0 + s, offAB, dA);
    ASYNC_LDS_LOAD_NT(ldsB0 + s, offAB, dBu);
    ASYNC_LDS_LOAD   (ldsC0 + s, offC,  Cm);
    if (j < L - 1) { offAB += stepAB; offC += 64u; }   // clamp for tiny L
  }

  // ---- main region: unconditional prefetch advance, i == 0 (mod 8) ------
  // Valid while every prefetch target j = i+k+PRE < L-1 for k in [0,8).
  const int mainEnd = L - 1 - PRE;
  int i = 0;
  for (; i + 8 <= mainEnd; i += 8) {
#pragma unroll
    for (int k = 0; k < 8; ++k) {
      // write stage (i+k+PRE)&7 == (k+PRE)&7 : compile-time constant
      const uint32_t sw = (uint32_t)((k + PRE) & (STAGES - 1)) * stageBytes;
      ASYNC_LDS_LOAD_NT(ldsA0 + sw, offAB, dA);
      ASYNC_LDS_LOAD_NT(ldsB0 + sw, offAB, dBu);
      ASYNC_LDS_LOAD   (ldsC0 + sw, offC,  Cm);
      offAB += stepAB;
      offC  += 64u;

      // 3 loads/stage * 7 stages in flight = 21 outstanding after issue;
      // wait <= 18 => oldest stage (timestep i+k) fully landed in LDS.
      asm volatile("s_wait_asynccnt 18" ::: "memory");

      const int sr = k << 8;                           // constant stage base
      const float a = sA[sr + t];
      const float b = sB[sr + t];
      const float c = sC[sr + t];

      xs = fmaf(a, xs, b);
      float p = xs * c;
      p += __shfl_xor(p, 8, 32);
      p += __shfl_xor(p, 4, 32);
      p += __shfl_xor(p, 2, 32);
      p += __shfl_xor(p, 1, 32);
      if (n == 0) out[(i + k) * D + dIdx] = p;
    }
  }

  // ---- tail: generic body with clamped prefetch (<= PRE+8 iterations) ----
  for (; i < L; ++i) {
    const uint32_t sw = (uint32_t)((i + PRE) & (STAGES - 1)) * stageBytes;
    ASYNC_LDS_LOAD_NT(ldsA0 + sw, offAB, dA);
    ASYNC_LDS_LOAD_NT(ldsB0 + sw, offAB, dBu);
    ASYNC_LDS_LOAD   (ldsC0 + sw, offC,  Cm);
    if (i + PRE < L - 1) { offAB += stepAB; offC += 64u; }

    asm volatile("s_wait_asynccnt 18" ::: "memory");

    const int sr = (i & (STAGES - 1)) << 8;
    const float a = sA[sr + t];
    const float b = sB[sr + t];
    const float c = sC[sr + t];

    xs = fmaf(a, xs, b);
    float p = xs * c;
    p += __shfl_xor(p, 8, 32);
    p += __shfl_xor(p, 4, 32);
    p += __shfl_xor(p, 2, 32);
    p += __shfl_xor(p, 1, 32);
    if (n == 0) out[i * D + dIdx] = p;
  }
}

extern "C" void kernel_launch(void* const* d_in, const int* in_sizes, int n_in,
                              void* d_out, int out_size, void* d_ws, size_t ws_size,
                              hipStream_t stream) {
  // inputs: [0]=l (scalar, device), [1]=x (B,D,N), [2]=deltaA (B,L,D,N),
  //         [3]=deltaB_u (B,L,D,N), [4]=C (B,L,N)
  const float* x   = (const float*)d_in[1];
  const float* dA  = (const float*)d_in[2];
  const float* dBu = (const float*)d_in[3];
  const float* Cm  = (const float*)d_in[4];
  float*       out = (float*)d_out;

  const int DN = in_sizes[1];                       // D*N (B=1)
  const int L  = (int)((long long)in_sizes[2] / DN);
  const int blocks = DN / 256;                      // one thread per (d,n)

  SelectiveScan_async_kernel<<<blocks, 256, 0, stream>>>(x, dA, dBu, Cm, out, L, DN);
}